// Sampler_72619307040792
// MI455X (gfx1250) — compile-verified
//
#include <hip/hip_runtime.h>
#include <stdint.h>

// ---------------------------------------------------------------------------
// Gumbel-softmax relaxed top-k (iterative), CDNA5 / gfx1250 (MI455X).
//
//   att : [B=256, N=1024, M=4] f32   (m fastest-varying)
//   k   : 128
//   out : khot [B, N, M] f32
//
// Design:
//   * grid = B blocks, 256 threads (8 wave32 waves) per block.
//   * 16 KB att tile staged to LDS via GLOBAL_LOAD_ASYNC_TO_LDS_B128
//     (ASYNCcnt path), then all loop state lives in VGPRs.
//   * per-step softmax over N: thread-local 4-row partials ->
//     5x ds-permute shuffle reduce (wave32) -> 8-wave LDS combine.
//     Double-buffered reduce scratch => only 2 barriers per step.
// ---------------------------------------------------------------------------

#ifndef __has_builtin
#define __has_builtin(x) 0
#endif

#if __has_builtin(__builtin_amdgcn_global_load_async_to_lds_b128)
#define USE_ASYNC_LDS 1
#else
#define USE_ASYNC_LDS 0
#endif

#define NN 1024           // nodes per graph
#define NM 4              // subgraph samples
#define TPB 256           // threads per block (8 waves of 32)
#define RPT 4             // rows (nodes) per thread = NN / TPB
#define NWAVES (TPB / 32)

#define EPS_TINY 1.17549435082228750796873653722e-38f   // float32 tiny
#define INV_TEMP 2.0f                                   // 1 / 0.5
#define NEG_HUGE (-3.402823466e38f)

#if USE_ASYNC_LDS
// Pointee type per the compiler diagnostic: int __attribute__((vector_size(16)))
typedef int v4i_t __attribute__((vector_size(16)));
typedef __attribute__((address_space(1))) v4i_t* gptr4_t;  // global
typedef __attribute__((address_space(3))) v4i_t* lptr4_t;  // LDS

static __device__ __forceinline__ void wait_async_zero() {
#if __has_builtin(__builtin_amdgcn_s_wait_asynccnt)
  __builtin_amdgcn_s_wait_asynccnt(0);
#else
  asm volatile("s_wait_asynccnt 0" ::: "memory");
#endif
}
#endif

// Deterministic counter-based uniform in [0,1) (PCG-style hash).
static __device__ __forceinline__ float hash_u01(uint32_t idx) {
  uint32_t s = idx * 747796405u + 2891336453u;
  uint32_t w = ((s >> ((s >> 28u) + 4u)) ^ s) * 277803737u;
  w = (w >> 22u) ^ w;
  return (float)(w >> 8) * (1.0f / 16777216.0f);
}

__global__ __launch_bounds__(TPB) void sampler_topk_kernel(
    const float* __restrict__ att, const int* __restrict__ kptr,
    float* __restrict__ out) {
  __shared__ float tile[NN * NM];          // 16 KB staging for one graph
  __shared__ float redA[NWAVES * NM];      // max partials
  __shared__ float redB[NWAVES * NM];      // sum partials

  const int b = blockIdx.x;
  const int t = threadIdx.x;
  const int lane = t & 31;
  const int wave = t >> 5;
  const int ksel = *kptr;                  // 128 in the reference setup

  const float* gbase = att + (size_t)b * (NN * NM);

#if USE_ASYNC_LDS
  // ---- async copy of the 16 KB tile: 4 x (256 lanes x 16 B) ----
  {
#pragma unroll
    for (int i = 0; i < (NN * NM) / (TPB * 4); ++i) {  // 4 issues
      const int j = i * TPB + t;                        // float4 index
      __builtin_amdgcn_global_load_async_to_lds_b128(
          (gptr4_t)(gbase + (size_t)j * 4),
          (lptr4_t)(tile + j * 4),
          /*imm offset*/ 0, /*cpol*/ 0);
    }
    wait_async_zero();   // each wave drains its own ASYNCcnt ...
    __syncthreads();     // ... then the whole tile is visible to the block
  }
#endif

  // ---- per-thread state: rows n = r*256 + t, all 4 columns, in VGPRs ----
  float ag[RPT][NM], onehot[RPT][NM], khot[RPT][NM];
#pragma unroll
  for (int r = 0; r < RPT; ++r) {
    const int n = r * TPB + t;
#if USE_ASYNC_LDS
    const float4 v = *(const float4*)(tile + n * NM);
#else
    const float4 v = *(const float4*)(gbase + (size_t)n * NM);
#endif
    const float av[NM] = {v.x, v.y, v.z, v.w};
#pragma unroll
    for (int m = 0; m < NM; ++m) {
      const uint32_t idx = (uint32_t)(((b * NN + n) * NM) + m);
      const float u = hash_u01(idx);
      // gumbel keys: ag = att - log(-log(u + tiny))
      ag[r][m] = av[m] - __logf(-__logf(u + EPS_TINY));
      onehot[r][m] = 0.0f;
      khot[r][m] = 0.0f;
    }
  }

  // ---- k sequential relaxed selections ----
  for (int it = 0; it < ksel; ++it) {
    // 1) ag += log(max(1 - onehot, tiny)); thread-local column maxima
    float cmax[NM];
#pragma unroll
    for (int m = 0; m < NM; ++m) cmax[m] = NEG_HUGE;
#pragma unroll
    for (int r = 0; r < RPT; ++r)
#pragma unroll
      for (int m = 0; m < NM; ++m) {
        const float mask = fmaxf(1.0f - onehot[r][m], EPS_TINY);
        ag[r][m] += __logf(mask);
        cmax[m] = fmaxf(cmax[m], ag[r][m]);
      }

    // 2) wave32 xor-shuffle max reduce (ds permute network)
#pragma unroll
    for (int m = 0; m < NM; ++m) {
#pragma unroll
      for (int off = 16; off >= 1; off >>= 1)
        cmax[m] = fmaxf(cmax[m], __shfl_xor(cmax[m], off, 32));
    }
    if (lane == 0) {
#pragma unroll
      for (int m = 0; m < NM; ++m) redA[wave * NM + m] = cmax[m];
    }
    __syncthreads();  // barrier #1
#pragma unroll
    for (int m = 0; m < NM; ++m) {
      float v = NEG_HUGE;
#pragma unroll
      for (int w = 0; w < NWAVES; ++w) v = fmaxf(v, redA[w * NM + m]);
      cmax[m] = v;
    }

    // 3) exp((ag - max) / T) and column sums
    float e[RPT][NM];
    float csum[NM] = {0.0f, 0.0f, 0.0f, 0.0f};
#pragma unroll
    for (int r = 0; r < RPT; ++r)
#pragma unroll
      for (int m = 0; m < NM; ++m) {
        const float x = (ag[r][m] - cmax[m]) * INV_TEMP;
        e[r][m] = __expf(x);
        csum[m] += e[r][m];
      }
#pragma unroll
    for (int m = 0; m < NM; ++m) {
#pragma unroll
      for (int off = 16; off >= 1; off >>= 1)
        csum[m] += __shfl_xor(csum[m], off, 32);
    }
    if (lane == 0) {
#pragma unroll
      for (int m = 0; m < NM; ++m) redB[wave * NM + m] = csum[m];
    }
    __syncthreads();  // barrier #2
#pragma unroll
    for (int m = 0; m < NM; ++m) {
      float v = 0.0f;
#pragma unroll
      for (int w = 0; w < NWAVES; ++w) v += redB[w * NM + m];
      csum[m] = v;
    }

    // 4) onehot = softmax; khot += onehot   (stays in VGPRs)
#pragma unroll
    for (int m = 0; m < NM; ++m) {
      const float inv = 1.0f / csum[m];
#pragma unroll
      for (int r = 0; r < RPT; ++r) {
        const float oh = e[r][m] * inv;
        onehot[r][m] = oh;
        khot[r][m] += oh;
      }
    }
    // redA reuse next iteration is fenced by barrier #2 of this iteration,
    // redB reuse by barrier #1 of the next iteration (double-buffered).
  }

  // ---- coalesced b128 store of khot ----
  float* obase = out + (size_t)b * (NN * NM);
#pragma unroll
  for (int r = 0; r < RPT; ++r) {
    const int n = r * TPB + t;
    float4 v;
    v.x = khot[r][0];
    v.y = khot[r][1];
    v.z = khot[r][2];
    v.w = khot[r][3];
    *(float4*)(obase + (size_t)n * NM) = v;
  }
}

extern "C" void kernel_launch(void* const* d_in, const int* in_sizes, int n_in,
                              void* d_out, int out_size, void* d_ws,
                              size_t ws_size, hipStream_t stream) {
  const float* att = (const float*)d_in[0];
  const int* kptr = (const int*)d_in[1];  // k (=128) as 1-element device int
  float* out = (float*)d_out;

  const int nb = in_sizes[0] / (NN * NM);  // number of graphs (256)
  sampler_topk_kernel<<<nb, TPB, 0, stream>>>(att, kptr, out);
  (void)n_in;
  (void)out_size;
  (void)d_ws;
  (void)ws_size;
}